// OnlineLabelSmoothing_18210661335666
// MI455X (gfx1250) — compile-verified
//
#include <hip/hip_runtime.h>
#include <hip/hip_bf16.h>

#define NC 8192      // classes
#define NB 16384     // batch
#define BT 256       // threads per row-block
#define CH 256       // pass-2 columns (c) per tile
#define TW 32        // pass-2 classes (t) per tile

typedef float v2f __attribute__((ext_vector_type(2)));
typedef float v8f __attribute__((ext_vector_type(8)));

// ---------------------------------------------------------------------------
// CDNA5 async global->LDS copy (ASYNCcnt path, bypasses VGPRs).
// INST_OFFSET is added to BOTH the global address and the LDS address.
// ---------------------------------------------------------------------------
#define ASYNC_B128(ldsaddr, gaddr, KOFF)                                        \
    asm volatile("global_load_async_to_lds_b128 %0, %1, off offset:" #KOFF      \
                 :: "v"(ldsaddr), "v"(gaddr) : "memory")

__device__ __forceinline__ void wait_async0() {
    asm volatile("s_wait_asynccnt 0x0" ::: "memory");
}

// ---------------------------------------------------------------------------
// Ones-matmul 64->16 reduction on the matrix pipe.
// D[m][n] = sum_k B[k][n]  (A = ones, C = 0). Correct for any bijective
// lane<->element mapping of B. Lane j<16, D element 0 = column sum j.
// EXEC must be all-ones at the call site (whole block converged).
// ---------------------------------------------------------------------------
__device__ __forceinline__ float wmma_colsum(float v0, float v1) {
    v2f A = {1.0f, 1.0f};
    v2f B = {v0, v1};
    v8f C = {};
    v8f D = __builtin_amdgcn_wmma_f32_16x16x4_f32(
        /*neg_a=*/false, A, /*neg_b=*/false, B,
        /*c_mod=*/(short)0, C, /*reuse_a=*/false, /*reuse_b=*/false);
    return D[0];
}

// ---------------------------------------------------------------------------
// Kernel A: zero the per-class integer counters (counts + cursor).
// ---------------------------------------------------------------------------
__global__ void ols_zero_counts(int* __restrict__ counts, int* __restrict__ cursor) {
    int i = blockIdx.x * blockDim.x + threadIdx.x;
    if (i < NC) { counts[i] = 0; cursor[i] = 0; }
}

// ---------------------------------------------------------------------------
// Kernel B (pass 1): one block per batch row.
//   async-DMA the 32KB row into LDS; reduce max/argmax (first-index tiebreak,
//   matches jnp.argmax) + sum(x) via LDS tree; sum(exp(x-max)) via the WMMA
//   ones-matmul reduction (3 barriers instead of 8). Emits per-row stats,
//   loss, and exact integer per-class hit counts.
// ---------------------------------------------------------------------------
__launch_bounds__(BT)
__global__ void ols_row_kernel(const float* __restrict__ outputs,
                               const int*   __restrict__ target,
                               const float* __restrict__ supervise,
                               float* __restrict__ m_arr,    // [NB] row max
                               float* __restrict__ q_arr,    // [NB] 1/Z
                               int*   __restrict__ hitcls,   // [NB] t if hit else -1
                               int*   __restrict__ counts,   // [NC] hit counts
                               float* __restrict__ loss_ws)  // [NB] per-row losses
{
    __shared__ __align__(16) float row[NC];
    __shared__ float s_f[BT];
    __shared__ int   s_i[BT];
    __shared__ float s_g[BT];
    __shared__ float s_w[16];

    const int tid = threadIdx.x;
    const int b   = blockIdx.x;
    const char* src = (const char*)(outputs + (size_t)b * NC);

    // ---- async copy: 8 x b128 per thread, lane-contiguous 16B chunks -------
    unsigned ldsAddr = (unsigned)(size_t)(&row[0]) + (unsigned)tid * 16u;
    const char* g    = src + (size_t)tid * 16u;
    ASYNC_B128(ldsAddr, g, 0);
    ASYNC_B128(ldsAddr, g, 4096);
    ASYNC_B128(ldsAddr, g, 8192);
    ASYNC_B128(ldsAddr, g, 12288);
    ASYNC_B128(ldsAddr, g, 16384);
    ASYNC_B128(ldsAddr, g, 20480);
    ASYNC_B128(ldsAddr, g, 24576);
    ASYNC_B128(ldsAddr, g, 28672);
    wait_async0();
    __syncthreads();

    // ---- local max/argmax (ascending idx => first occurrence) + sum(x) ----
    float lmax = -3.402823466e38f;
    int   lidx = 0;
    float lsum = 0.0f;
#pragma unroll
    for (int k = 0; k < NC / BT; ++k) {
        int idx = k * BT + tid;
        float v = row[idx];
        lsum += v;
        if (v > lmax) { lmax = v; lidx = idx; }
    }
    s_f[tid] = lmax; s_i[tid] = lidx; s_g[tid] = lsum;
    __syncthreads();
    for (int s = BT / 2; s > 0; s >>= 1) {
        if (tid < s) {
            float ov = s_f[tid + s]; int oi = s_i[tid + s];
            if (ov > s_f[tid] || (ov == s_f[tid] && oi < s_i[tid])) {
                s_f[tid] = ov; s_i[tid] = oi;
            }
            s_g[tid] += s_g[tid + s];
        }
        __syncthreads();
    }
    const float m    = s_f[0];
    const int   amax = s_i[0];
    const float sumx = s_g[0];
    __syncthreads();   // before reusing s_g

    // ---- sum of exp(x - max): WMMA ones-matmul block reduction ------------
    float lexp = 0.0f;
#pragma unroll
    for (int k = 0; k < NC / BT; ++k) {
        int idx = k * BT + tid;
        lexp += expf(row[idx] - m);
    }
    s_g[tid] = lexp;
    __syncthreads();
    {
        int l = tid & 31;   // every wave computes the identical reduction: EXEC all-ones
        float v0 = s_g[l]      + s_g[l + 64]  + s_g[l + 128] + s_g[l + 192];
        float v1 = s_g[l + 32] + s_g[l + 96]  + s_g[l + 160] + s_g[l + 224];
        float d0 = wmma_colsum(v0, v1);
        if (tid < 16) s_w[tid] = d0;
    }
    __syncthreads();
    float Z = 0.0f;
#pragma unroll
    for (int i = 0; i < 16; ++i) Z += s_w[i];   // identical in all threads

    if (tid == 0) {
        const float logZ = logf(Z);
        const int   t    = target[b];
        const bool  hit  = (amax == t);

        m_arr[b]  = m;
        q_arr[b]  = 1.0f / Z;
        hitcls[b] = hit ? t : -1;
        if (hit) atomicAdd(&counts[t], 1);

        float diag = supervise[0];   // 1 - smoothing
        float off  = supervise[1];   // smoothing / (C-1)
        float logp_t  = row[t] - m - logZ;
        float sumlogp = sumx - (float)NC * (m + logZ);
        float hard = -logp_t;
        float soft = -(off * (sumlogp - logp_t) + diag * logp_t);
        loss_ws[b] = (0.5f * hard + 0.5f * soft) * (1.0f / (float)NB);
    }
}

// ---------------------------------------------------------------------------
// Kernel C: exclusive prefix scan of counts -> offs (NC+1 entries), and the
// exact idx_count_new output (no fp atomics anywhere).
// ---------------------------------------------------------------------------
__global__ void ols_scan_kernel(const int* __restrict__ counts,
                                int* __restrict__ offs,            // [NC+1]
                                const float* __restrict__ idx_in,  // [NC]
                                float* __restrict__ idx_out) {     // [NC]
    __shared__ int part[256];
    const int tid  = threadIdx.x;
    const int base = tid * (NC / 256);
    int s = 0;
#pragma unroll
    for (int i = 0; i < NC / 256; ++i) s += counts[base + i];
    part[tid] = s;
    __syncthreads();
    if (tid == 0) {
        int r = 0;
        for (int j = 0; j < 256; ++j) { int v = part[j]; part[j] = r; r += v; }
    }
    __syncthreads();
    int run = part[tid];
#pragma unroll
    for (int i = 0; i < NC / 256; ++i) {
        int idx = base + i;
        int c = counts[idx];
        offs[idx] = run;
        idx_out[idx] = idx_in[idx] + (float)c;
        run += c;
    }
    if (tid == 255) offs[NC] = run;
}

// ---------------------------------------------------------------------------
// Kernel D: bucket hit rows into per-class contiguous lists.
// ---------------------------------------------------------------------------
__global__ void ols_fill_kernel(const int* __restrict__ hitcls,
                                const int* __restrict__ offs,
                                int* __restrict__ cursor,
                                int* __restrict__ list) {
    int b = blockIdx.x * blockDim.x + threadIdx.x;
    if (b >= NB) return;
    int t = hitcls[b];
    if (t >= 0) {
        int pos = atomicAdd(&cursor[t], 1);
        list[offs[t] + pos] = b;
    }
}

// ---------------------------------------------------------------------------
// Kernel E (pass 2): tile = TW classes x CH columns. Replays only the hit
// rows of its class range (contiguous list slice), accumulates probs into a
// padded LDS tile (bank-conflict-free, no atomics), then writes
//   out[c*NC + t] = update[c*NC + t] + acc[c][t]
// exactly once per element, coalesced along t.
// ---------------------------------------------------------------------------
__launch_bounds__(CH)
__global__ void ols_scatter_tile(const float* __restrict__ outputs,
                                 const float* __restrict__ upd_in,
                                 const float* __restrict__ m_arr,
                                 const float* __restrict__ q_arr,
                                 const int*   __restrict__ hitcls,
                                 const int*   __restrict__ offs,
                                 const int*   __restrict__ list,
                                 float* __restrict__ out_u) {
    __shared__ float accT[CH][TW + 1];   // +1 pad: lane stride 33 words -> no bank conflicts
    const int tid = threadIdx.x;
    const int c0  = blockIdx.x * CH;
    const int t0  = blockIdx.y * TW;

    for (int i = tid; i < CH * (TW + 1); i += CH) ((float*)accT)[i] = 0.0f;
    __syncthreads();

    const int rs = offs[t0];
    const int re = offs[t0 + TW];

    int i = rs;
    for (; i + 1 < re; i += 2) {            // light 2x unroll to hide load latency
        int b0 = list[i],     b1 = list[i + 1];
        int u0 = hitcls[b0],  u1 = hitcls[b1];
        float x0 = outputs[(size_t)b0 * NC + c0 + tid];
        float x1 = outputs[(size_t)b1 * NC + c0 + tid];
        float p0 = expf(x0 - m_arr[b0]) * q_arr[b0];
        float p1 = expf(x1 - m_arr[b1]) * q_arr[b1];
        accT[tid][u0 - t0] += p0;
        accT[tid][u1 - t0] += p1;
    }
    if (i < re) {
        int b0 = list[i];
        int u0 = hitcls[b0];
        float x0 = outputs[(size_t)b0 * NC + c0 + tid];
        accT[tid][u0 - t0] += expf(x0 - m_arr[b0]) * q_arr[b0];
    }
    __syncthreads();

#pragma unroll
    for (int idx0 = 0; idx0 < CH * TW; idx0 += CH) {
        int idx = idx0 + tid;
        int tl = idx & (TW - 1);
        int cl = idx / TW;
        size_t o = (size_t)(c0 + cl) * NC + (t0 + tl);
        out_u[o] = upd_in[o] + accT[cl][tl];
    }
}

// ---------------------------------------------------------------------------
// Kernel F: deterministic loss reduction; final 256->1 uses the WMMA
// ones-matmul reduction.
// ---------------------------------------------------------------------------
__global__ void ols_loss_reduce(const float* __restrict__ ws,
                                float* __restrict__ out) {
    __shared__ float s[256];
    __shared__ float fin[16];
    int tid = threadIdx.x;
    float a = 0.0f;
    for (int k = tid; k < NB; k += 256) a += ws[k];
    s[tid] = a;
    __syncthreads();
    int l = tid & 31;
    float v0 = s[l]      + s[l + 64] + s[l + 128] + s[l + 192];
    float v1 = s[l + 32] + s[l + 96] + s[l + 160] + s[l + 224];
    float d0 = wmma_colsum(v0, v1);
    if (tid < 16) fin[tid] = d0;
    __syncthreads();
    if (tid == 0) {
        float r = 0.0f;
#pragma unroll
        for (int i = 0; i < 16; ++i) r += fin[i];
        out[0] = r;
    }
}

// ---------------------------------------------------------------------------
extern "C" void kernel_launch(void* const* d_in, const int* in_sizes, int n_in,
                              void* d_out, int out_size, void* d_ws, size_t ws_size,
                              hipStream_t stream) {
    const float* outputs   = (const float*)d_in[0];   // [NB, NC] f32
    const int*   target    = (const int*)  d_in[1];   // [NB] i32
    const float* supervise = (const float*)d_in[2];   // [NC, NC] f32
    const float* update    = (const float*)d_in[3];   // [NC, NC] f32
    const float* idx_count = (const float*)d_in[4];   // [NC] f32

    float* out     = (float*)d_out;                   // [0]=loss, [1..]=update_new, then idx_count_new
    float* out_u   = out + 1;
    float* idx_out = out + 1 + (size_t)NC * NC;

    // workspace layout (~420 KB)
    float* ws      = (float*)d_ws;
    float* loss_ws = ws;                               // NB floats
    float* m_arr   = ws + NB;                          // NB floats
    float* q_arr   = ws + 2 * (size_t)NB;              // NB floats
    int*   hitcls  = (int*)(ws + 3 * (size_t)NB);      // NB ints
    int*   counts  = hitcls + NB;                      // NC ints
    int*   offs    = counts + NC;                      // NC+1 ints
    int*   cursor  = offs + NC + 1;                    // NC ints
    int*   list    = cursor + NC;                      // NB ints

    // A: zero per-class counters
    ols_zero_counts<<<(NC + 255) / 256, 256, 0, stream>>>(counts, cursor);

    // B: pass 1 — per-row softmax stats via async-LDS streaming + WMMA reduce
    ols_row_kernel<<<NB, BT, 0, stream>>>(outputs, target, supervise,
                                          m_arr, q_arr, hitcls, counts, loss_ws);

    // C: prefix scan + exact idx_count output
    ols_scan_kernel<<<1, 256, 0, stream>>>(counts, offs, idx_count, idx_out);

    // D: bucket hit rows by class
    ols_fill_kernel<<<NB / 256, 256, 0, stream>>>(hitcls, offs, cursor, list);

    // E: pass 2 — grouped scatter, writes update_new exactly once per element
    dim3 sg(NC / CH, NC / TW);
    ols_scatter_tile<<<sg, CH, 0, stream>>>(outputs, update, m_arr, q_arr,
                                            hitcls, offs, list, out_u);

    // F: final loss
    ols_loss_reduce<<<1, 256, 0, stream>>>(loss_ws, out);
}